// GATEncoder_25933012533382
// MI455X (gfx1250) — compile-verified
//
#include <hip/hip_runtime.h>

#define SLOPE 0.2f

constexpr int NN    = 50000;
constexpr int EE    = 400000;
constexpr int INF_  = 128;
constexpr int HID   = 64;
constexpr int HEADS = 4;
constexpr int OUTD  = 32;

typedef __attribute__((ext_vector_type(2))) float v2f;
typedef __attribute__((ext_vector_type(8))) float v8f;

// ---------------------------------------------------------------------------
// GEMM: C(M x Ncols) = A(M x K, row major) * B(Ncols x K, row major)^T
// One wave32 per 16x16 output tile, V_WMMA_F32_16X16X4_F32, K stepped by 4.
// A 16x4 layout: lane l holds row m=l%16, K = 2*(l/16) + vgpr.
// B 4x16 layout: lane l holds col n=l%16, K = 2*(l/16) + vgpr.
// C/D: VGPR j -> (row = j + 8*(l/16), col = l%16).
// ---------------------------------------------------------------------------
__global__ void gemm_wmma_nt(const float* __restrict__ A,
                             const float* __restrict__ B,
                             float* __restrict__ C,
                             int K, int Ncols) {
    const int tile_r = blockIdx.x;
    const int tile_c = blockIdx.y;
    const int lane   = threadIdx.x & 31;
    const int idx16  = lane & 15;
    const int half   = lane >> 4;

    const float* arow = A + (long)(tile_r * 16 + idx16) * K + 2 * half;
    const float* brow = B + (long)(tile_c * 16 + idx16) * K + 2 * half;

    v8f acc = {};
    for (int k = 0; k < K; k += 4) {
        v2f a = *(const v2f*)(arow + k);
        v2f b = *(const v2f*)(brow + k);
        acc = __builtin_amdgcn_wmma_f32_16x16x4_f32(
            false, a, false, b, (short)0, acc, false, false);
    }

    const int row0 = tile_r * 16 + half * 8;
    const int col  = tile_c * 16 + idx16;
    float* cp = C + (long)row0 * Ncols + col;
#pragma unroll
    for (int j = 0; j < 8; ++j) cp[(long)j * Ncols] = acc[j];
}

// el[i,h] = dot(z[i,h,:], al[h,:]);  er likewise. One thread per (i,h).
__global__ void attn_lr_kernel(const float* __restrict__ z,
                               const float* __restrict__ al,
                               const float* __restrict__ ar,
                               float* __restrict__ el,
                               float* __restrict__ er,
                               int n, int H, int D) {
    int g = blockIdx.x * blockDim.x + threadIdx.x;
    if (g >= n * H) return;
    int i = g / H, h = g % H;
    const float* zr = z + (long)i * H * D + h * D;
    float sl = 0.f, sr = 0.f;
    for (int d = 0; d < D; ++d) {
        float v = zr[d];
        sl += v * al[h * D + d];
        sr += v * ar[h * D + d];
    }
    el[g] = sl;
    er[g] = sr;
}

__global__ void zero_int_kernel(int* __restrict__ p, int n) {
    int i = blockIdx.x * blockDim.x + threadIdx.x;
    if (i < n) p[i] = 0;
}

__global__ void copy_int_kernel(const int* __restrict__ a, int* __restrict__ b, int n) {
    int i = blockIdx.x * blockDim.x + threadIdx.x;
    if (i < n) b[i] = a[i];
}

__global__ void count_deg_kernel(const int* __restrict__ dst, int* __restrict__ deg, int e) {
    int i = blockIdx.x * blockDim.x + threadIdx.x;
    if (i < e) atomicAdd(&deg[dst[i]], 1);
}

// Single-block exclusive scan: rowptr[0]=0, rowptr[i+1]=sum(deg[0..i]).
__global__ void scan_kernel(const int* __restrict__ deg, int* __restrict__ rowptr, int n) {
    __shared__ int buf[1024];
    __shared__ int carry_s;
    const int tid = threadIdx.x;
    if (tid == 0) { carry_s = 0; rowptr[0] = 0; }
    __syncthreads();
    for (int base = 0; base < n; base += 1024) {
        int i = base + tid;
        buf[tid] = (i < n) ? deg[i] : 0;
        __syncthreads();
        for (int off = 1; off < 1024; off <<= 1) {
            int t = (tid >= off) ? buf[tid - off] : 0;
            __syncthreads();
            buf[tid] += t;
            __syncthreads();
        }
        if (i < n) rowptr[i + 1] = carry_s + buf[tid];
        __syncthreads();
        if (tid == 0) carry_s += buf[1023];
        __syncthreads();
    }
}

__global__ void scatter_kernel(const int* __restrict__ dst, int* __restrict__ cursor,
                               int* __restrict__ eids, int e) {
    int i = blockIdx.x * blockDim.x + threadIdx.x;
    if (i < e) {
        int p = atomicAdd(&cursor[dst[i]], 1);
        eids[p] = i;
    }
}

// Sort edge ids within each node's segment -> deterministic accumulation order.
__global__ void sort_eids_kernel(const int* __restrict__ rowptr, int* __restrict__ eids, int n) {
    int i = blockIdx.x * blockDim.x + threadIdx.x;
    if (i >= n) return;
    int s = rowptr[i], e = rowptr[i + 1];
    for (int a = s + 1; a < e; ++a) {
        int key = eids[a];
        int b = a - 1;
        while (b >= s && eids[b] > key) { eids[b + 1] = eids[b]; --b; }
        eids[b + 1] = key;
    }
}

__device__ __forceinline__ float lrelu(float v) { return v > 0.f ? v : SLOPE * v; }

// One wave32 per destination node: segment softmax + weighted aggregation.
template <int H, int D, bool RELU>
__global__ void gat_aggregate(const float* __restrict__ z,
                              const float* __restrict__ el,
                              const float* __restrict__ er,
                              const float* __restrict__ bias,
                              const int* __restrict__ srcv,
                              const int* __restrict__ rowptr,
                              const int* __restrict__ eids,
                              float* __restrict__ out, int n) {
    constexpr int HD = H * D;
    constexpr int S  = HD / 32;  // output slots per lane (HD multiple of 32)
    const int lane = threadIdx.x & 31;
    const int wave = threadIdx.x >> 5;
    const int node = blockIdx.x * (blockDim.x >> 5) + wave;
    if (node >= n) return;

    const int s0 = rowptr[node], s1 = rowptr[node + 1];

    float erh[H];
#pragma unroll
    for (int h = 0; h < H; ++h) erh[h] = er[node * H + h];

    // pass 1: per-head max over incoming edges
    float mx[H];
#pragma unroll
    for (int h = 0; h < H; ++h) mx[h] = -3.4e38f;
    for (int eidx = s0 + lane; eidx < s1; eidx += 32) {
        int sN = srcv[eids[eidx]];
#pragma unroll
        for (int h = 0; h < H; ++h)
            mx[h] = fmaxf(mx[h], lrelu(el[sN * H + h] + erh[h]));
    }
#pragma unroll
    for (int off = 16; off > 0; off >>= 1)
#pragma unroll
        for (int h = 0; h < H; ++h)
            mx[h] = fmaxf(mx[h], __shfl_xor(mx[h], off, 32));

    // pass 2: softmax denominator
    float den[H];
#pragma unroll
    for (int h = 0; h < H; ++h) den[h] = 0.f;
    for (int eidx = s0 + lane; eidx < s1; eidx += 32) {
        int sN = srcv[eids[eidx]];
#pragma unroll
        for (int h = 0; h < H; ++h)
            den[h] += __expf(lrelu(el[sN * H + h] + erh[h]) - mx[h]);
    }
#pragma unroll
    for (int off = 16; off > 0; off >>= 1)
#pragma unroll
        for (int h = 0; h < H; ++h)
            den[h] += __shfl_xor(den[h], off, 32);

    // pass 3: sequential over edges (deterministic), lanes parallel over (h,d)
    float acc[S];
#pragma unroll
    for (int j = 0; j < S; ++j) acc[j] = 0.f;
    for (int eidx = s0; eidx < s1; ++eidx) {
        int sN = srcv[eids[eidx]];
        float alpha[H];
#pragma unroll
        for (int h = 0; h < H; ++h)
            alpha[h] = __expf(lrelu(el[sN * H + h] + erh[h]) - mx[h]) / den[h];
        const float* zr = z + (long)sN * HD;
#pragma unroll
        for (int j = 0; j < S; ++j) {
            int idx = lane + 32 * j;
            acc[j] += alpha[idx / D] * zr[idx];
        }
    }
#pragma unroll
    for (int j = 0; j < S; ++j) {
        int idx = lane + 32 * j;
        float v = acc[j] + bias[idx];
        if (RELU) v = fmaxf(v, 0.f);
        out[(long)node * HD + idx] = v;
    }
}

// Deterministic single-block mean over nodes: out[c] = mean(h2[:, c]).
__global__ void mean_nodes_kernel(const float* __restrict__ h2, float* __restrict__ out, int n) {
    __shared__ float red[1024];
    const int tid = threadIdx.x;   // 1024 threads
    const int c   = tid & 31;
    const int rg  = tid >> 5;      // 0..31
    float s = 0.f;
    for (int r = rg; r < n; r += 32) s += h2[(long)r * OUTD + c];
    red[tid] = s;
    __syncthreads();
    for (int off = 16; off > 0; off >>= 1) {
        if (rg < off) red[tid] += red[tid + off * 32];
        __syncthreads();
    }
    if (rg == 0) out[c] = red[c] * (1.0f / n);
}

extern "C" void kernel_launch(void* const* d_in, const int* in_sizes, int n_in,
                              void* d_out, int out_size, void* d_ws, size_t ws_size,
                              hipStream_t stream) {
    (void)in_sizes; (void)n_in; (void)out_size; (void)ws_size;

    const float* x   = (const float*)d_in[0];
    const int*   src = (const int*)d_in[1];
    const int*   dst = (const int*)d_in[2];
    const float* W1  = (const float*)d_in[3];
    const float* al1 = (const float*)d_in[4];
    const float* ar1 = (const float*)d_in[5];
    const float* b1  = (const float*)d_in[6];
    const float* W2  = (const float*)d_in[7];
    const float* al2 = (const float*)d_in[8];
    const float* ar2 = (const float*)d_in[9];
    const float* b2  = (const float*)d_in[10];
    float* out = (float*)d_out;

    // carve workspace (256B aligned chunks)
    char* ws = (char*)d_ws;
    auto carve = [&](size_t bytes) -> char* {
        char* p = ws;
        ws += (bytes + 255) & ~(size_t)255;
        return p;
    };
    float* z1     = (float*)carve((size_t)NN * HEADS * HID * 4);   // 50000 x 256
    float* h1     = (float*)carve((size_t)NN * HEADS * HID * 4);   // 50000 x 256
    float* el1    = (float*)carve((size_t)NN * HEADS * 4);
    float* er1    = (float*)carve((size_t)NN * HEADS * 4);
    float* z2     = (float*)carve((size_t)NN * OUTD * 4);          // 50000 x 32
    float* h2     = (float*)carve((size_t)NN * OUTD * 4);
    float* el2    = (float*)carve((size_t)NN * 4);
    float* er2    = (float*)carve((size_t)NN * 4);
    int*   deg    = (int*)carve((size_t)NN * 4);
    int*   rowptr = (int*)carve((size_t)(NN + 1) * 4);
    int*   cursor = (int*)carve((size_t)NN * 4);
    int*   eids   = (int*)carve((size_t)EE * 4);

    const int T = 256;
    const int gN  = (NN + T - 1) / T;
    const int gE  = (EE + T - 1) / T;

    // ---- build CSR by dst (deterministic after sort) ----
    zero_int_kernel<<<gN, T, 0, stream>>>(deg, NN);
    count_deg_kernel<<<gE, T, 0, stream>>>(dst, deg, EE);
    scan_kernel<<<1, 1024, 0, stream>>>(deg, rowptr, NN);
    copy_int_kernel<<<gN, T, 0, stream>>>(rowptr, cursor, NN);
    scatter_kernel<<<gE, T, 0, stream>>>(dst, cursor, eids, EE);
    sort_eids_kernel<<<gN, T, 0, stream>>>(rowptr, eids, NN);

    // ---- layer 1: z1 = x @ W1^T  (50000x128 * 128x256) ----
    gemm_wmma_nt<<<dim3(NN / 16, (HEADS * HID) / 16), 32, 0, stream>>>(
        x, W1, z1, INF_, HEADS * HID);
    attn_lr_kernel<<<(NN * HEADS + T - 1) / T, T, 0, stream>>>(
        z1, al1, ar1, el1, er1, NN, HEADS, HID);
    gat_aggregate<HEADS, HID, true><<<NN / 8, 256, 0, stream>>>(
        z1, el1, er1, b1, src, rowptr, eids, h1, NN);

    // ---- layer 2: z2 = h1 @ W2^T  (50000x256 * 256x32) ----
    gemm_wmma_nt<<<dim3(NN / 16, OUTD / 16), 32, 0, stream>>>(
        h1, W2, z2, HEADS * HID, OUTD);
    attn_lr_kernel<<<(NN + T - 1) / T, T, 0, stream>>>(
        z2, al2, ar2, el2, er2, NN, 1, OUTD);
    gat_aggregate<1, OUTD, false><<<NN / 8, 256, 0, stream>>>(
        z2, el2, er2, b2, src, rowptr, eids, h2, NN);

    // ---- mean over nodes ----
    mean_nodes_kernel<<<1, 1024, 0, stream>>>(h2, out, NN);
}